// MG_encoder_1357209665644
// MI455X (gfx1250) — compile-verified
//
#include <hip/hip_runtime.h>
#include <math.h>

#define NN  100000
#define NEI 1600000
#define NEC 800000
#define NG  2000
#define DM  128      // all hidden widths are 128 (NH*H1 = NH*H2 = EMB)

typedef float v2f __attribute__((ext_vector_type(2)));
typedef float v8f __attribute__((ext_vector_type(8)));

#if defined(__HIP_DEVICE_COMPILE__)
#if !__has_builtin(__builtin_amdgcn_wmma_f32_16x16x4_f32)
#error "missing __builtin_amdgcn_wmma_f32_16x16x4_f32 on this target"
#endif
#endif

__device__ __forceinline__ float lrelu(float x) { return x >= 0.f ? x : 0.2f * x; }

// ---------------------------------------------------------------------------
// GEMM: OUT[nrows,128] = X[nrows,K] @ W[K,128], optional per-row output scale.
// Block = 256 threads = 8 waves; block computes a 128x128 panel; wave w owns
// rows [m0, m0+16) and all 8 N-tiles. f32 WMMA 16x16x4.
// A frag (16x4 f32): lanes 0-15 hold (K=k0,k0+1) of row M=lane,
//                    lanes 16-31 hold (K=k0+2,k0+3) of row M=lane-16.
// B frag (4x16 f32): VGPR0 = row k0 (lanes 0-15) / row k0+2 (lanes 16-31),
//                    VGPR1 = row k0+1 / k0+3, cols = n0 + (lane&15).
// ---------------------------------------------------------------------------
template <int K>
__global__ __launch_bounds__(256) void k_gemm_wmma(
    const float* __restrict__ X, const float* __restrict__ W,
    float* __restrict__ OUT, const float* __restrict__ rowscale, int nrows) {
  const int wid  = threadIdx.x >> 5;
  const int lane = threadIdx.x & 31;
  const int half = lane >> 4;
  const int l16  = lane & 15;
  const int m0   = blockIdx.x * 128 + wid * 16;

  int arow = m0 + l16;
  if (arow >= nrows) arow = nrows - 1;  // clamp loads, mask stores
  const float* xrow = X + (size_t)arow * K + 2 * half;
  const float* wcol = W + l16;

  v8f acc[8];
#pragma unroll
  for (int t = 0; t < 8; t++) { v8f z = {}; acc[t] = z; }

  for (int k0 = 0; k0 < K; k0 += 4) {
    v2f a = *(const v2f*)(xrow + k0);
    const float* wr = wcol + (size_t)(k0 + 2 * half) * DM;
#pragma unroll
    for (int t = 0; t < 8; t++) {
      v2f b;
      b.x = wr[t * 16];
      b.y = wr[DM + t * 16];
      acc[t] = __builtin_amdgcn_wmma_f32_16x16x4_f32(
          false, a, false, b, (short)0, acc[t], false, false);
    }
  }

  // C/D layout: VGPR j: lanes 0-15 -> M=j, lanes 16-31 -> M=j+8; N = t*16+l16
  float sc[8];
#pragma unroll
  for (int j = 0; j < 8; j++) {
    int r = m0 + 8 * half + j;
    sc[j] = (rowscale != nullptr && r < nrows) ? rowscale[r] : 1.f;
  }
#pragma unroll
  for (int t = 0; t < 8; t++) {
#pragma unroll
    for (int j = 0; j < 8; j++) {
      int r = m0 + 8 * half + j;
      if (r < nrows) OUT[(size_t)r * DM + t * 16 + l16] = acc[t][j] * sc[j];
    }
  }
}

// ------------------------- degree counting / CSR build ----------------------
__global__ void k_count(const int* __restrict__ src, const int* __restrict__ dst,
                        int* cnt_src, int* __restrict__ cnt_dst, int E) {
  int e = blockIdx.x * blockDim.x + threadIdx.x;
  if (e < E) {
    if (cnt_src) atomicAdd(&cnt_src[src[e]], 1);
    atomicAdd(&cnt_dst[dst[e]], 1);
  }
}

__global__ void k_deginv(const int* __restrict__ cnt, float* __restrict__ inv, int n) {
  int i = blockIdx.x * blockDim.x + threadIdx.x;
  if (i < n) inv[i] = 1.f / sqrtf(fmaxf((float)cnt[i], 1.f));
}

__global__ void k_scan_block(const int* __restrict__ cnt, int* __restrict__ excl,
                             int* __restrict__ bsum, int n) {
  __shared__ int s[256];
  int i = blockIdx.x * 256 + threadIdx.x;
  int v = (i < n) ? cnt[i] : 0;
  s[threadIdx.x] = v;
  __syncthreads();
  for (int off = 1; off < 256; off <<= 1) {
    int t = (threadIdx.x >= off) ? s[threadIdx.x - off] : 0;
    __syncthreads();
    s[threadIdx.x] += t;
    __syncthreads();
  }
  if (i < n) excl[i] = s[threadIdx.x] - v;
  if (threadIdx.x == 255) bsum[blockIdx.x] = s[255];
}

__global__ void k_scan_single(int* __restrict__ bsum, int nb) {
  __shared__ int s[1024];
  int v = (threadIdx.x < nb) ? bsum[threadIdx.x] : 0;
  s[threadIdx.x] = v;
  __syncthreads();
  for (int off = 1; off < 1024; off <<= 1) {
    int t = (threadIdx.x >= off) ? s[threadIdx.x - off] : 0;
    __syncthreads();
    s[threadIdx.x] += t;
    __syncthreads();
  }
  if (threadIdx.x < nb) bsum[threadIdx.x] = s[threadIdx.x] - v;  // exclusive
}

__global__ void k_scan_add(int* __restrict__ excl, const int* __restrict__ boff,
                           int* __restrict__ cur, int n) {
  int i = blockIdx.x * 256 + threadIdx.x;
  if (i < n) {
    int r = excl[i] + boff[blockIdx.x];
    excl[i] = r;
    cur[i]  = r;
  }
}

__global__ void k_csr_fill(const int* __restrict__ src, const int* __restrict__ dst,
                           int* __restrict__ cur, int* __restrict__ entries, int E) {
  int e = blockIdx.x * blockDim.x + threadIdx.x;
  if (e < E) {
    int p = atomicAdd(&cur[dst[e]], 1);
    entries[p] = src[e];
  }
}

// --------------------- GraphConv aggregation (gather) -----------------------
__global__ __launch_bounds__(256) void k_gc_gather(
    const float* __restrict__ xw, const int* __restrict__ rowptr,
    const int* __restrict__ cnt, const int* __restrict__ entries,
    float* __restrict__ agg, int n) {
  int w    = (blockIdx.x * blockDim.x + threadIdx.x) >> 5;
  int lane = threadIdx.x & 31;
  if (w >= n) return;
  int s0 = rowptr[w], c = cnt[w];
  float a0 = 0.f, a1 = 0.f, a2 = 0.f, a3 = 0.f;
  if (c > 0) {
    int s = entries[s0];
    for (int j = 0; j < c; j++) {
      int nx = (j + 1 < c) ? entries[s0 + j + 1] : s;
      __builtin_prefetch(xw + (size_t)nx * DM + lane * 4, 0, 1);
      const float4 v = *(const float4*)(xw + (size_t)s * DM + lane * 4);
      a0 += v.x; a1 += v.y; a2 += v.z; a3 += v.w;
      s = nx;
    }
  }
  *(float4*)(agg + (size_t)w * DM + lane * 4) = make_float4(a0, a1, a2, a3);
}

// ------------------------------ GAT pieces ----------------------------------
template <int H>
__global__ void k_eler(const float* __restrict__ f, const float* __restrict__ al,
                       const float* __restrict__ ar, float* __restrict__ el,
                       float* __restrict__ er, int n) {
  const int d = DM / H;
  int t = blockIdx.x * blockDim.x + threadIdx.x;
  if (t >= n * H) return;
  int i = t / H, h = t - i * H;
  const float* fr  = f + (size_t)i * DM + h * d;
  const float* alh = al + h * d;
  const float* arh = ar + h * d;
  float a = 0.f, b = 0.f;
  for (int c = 0; c < d; c++) { float v = fr[c]; a += v * alh[c]; b += v * arh[c]; }
  el[t] = a;
  er[t] = b;
}

template <int H>
__global__ __launch_bounds__(256) void k_gat_gather(
    const float* __restrict__ f, const float* __restrict__ el,
    const float* __restrict__ er, const int* __restrict__ rowptr,
    const int* __restrict__ cnt, const int* __restrict__ entries,
    float* __restrict__ outb, float* __restrict__ attsum, int n) {
  int w    = (blockIdx.x * blockDim.x + threadIdx.x) >> 5;
  int lane = threadIdx.x & 31;
  if (w >= n) return;
  const int d = DM / H;
  const int h = (lane * 4) / d;  // head owning this lane's 4 columns
  int s0 = rowptr[w], c = cnt[w];
  float erv = er[(size_t)w * H + h];
  float m = -3.0e38f;
  for (int j = 0; j < c; j++) {
    int s = entries[s0 + j];
    m = fmaxf(m, lrelu(el[(size_t)s * H + h] + erv));
  }
  float ssum = 0.f;
  for (int j = 0; j < c; j++) {
    int s = entries[s0 + j];
    ssum += expf(lrelu(el[(size_t)s * H + h] + erv) - m);
  }
  float a0 = 0.f, a1 = 0.f, a2 = 0.f, a3 = 0.f;
  for (int j = 0; j < c; j++) {
    int s = entries[s0 + j];
    float alpha = expf(lrelu(el[(size_t)s * H + h] + erv) - m) / ssum;
    __builtin_prefetch(f + (size_t)((j + 1 < c) ? entries[s0 + j + 1] : s) * DM + lane * 4, 0, 1);
    const float4 v = *(const float4*)(f + (size_t)s * DM + lane * 4);
    a0 += alpha * v.x; a1 += alpha * v.y; a2 += alpha * v.z; a3 += alpha * v.w;
    if (H == 1 && attsum != nullptr && lane == 0) atomicAdd(&attsum[s], alpha);
  }
  *(float4*)(outb + (size_t)w * DM + lane * 4) = make_float4(a0, a1, a2, a3);
}

// -------------------- combine + batchnorm + pooling -------------------------
__global__ __launch_bounds__(128) void k_combine(
    const float* __restrict__ agg, const float* __restrict__ dininv,
    const float* __restrict__ bgc, const float* __restrict__ hc,
    const float* __restrict__ bgat, float* __restrict__ z,
    float* __restrict__ bsum, float* __restrict__ bsq, int n, int act) {
  int c = threadIdx.x;
  float bc = bgc[c] + bgat[c];
  float s = 0.f, s2 = 0.f;
  for (int i = blockIdx.x; i < n; i += gridDim.x) {
    size_t idx = (size_t)i * DM + c;
    float v = agg[idx] * dininv[i] + hc[idx] + bc;
    if (act) v = lrelu(v);
    z[idx] = v;
    s += v;
    s2 += v * v;
  }
  atomicAdd(&bsum[c], s);
  atomicAdd(&bsq[c], s2);
}

__global__ void k_bnfinal(const float* __restrict__ bsum, const float* __restrict__ bsq,
                          const float* __restrict__ g, const float* __restrict__ be,
                          float* __restrict__ scale, float* __restrict__ shift) {
  int c = threadIdx.x;
  float invn = 1.f / (float)NN;
  float mu  = bsum[c] * invn;
  float var = bsq[c] * invn - mu * mu;
  float sc  = g[c] / sqrtf(var + 1e-5f);
  scale[c] = sc;
  shift[c] = be[c] - mu * sc;
}

__global__ void k_bnapply(const float* __restrict__ z, const float* __restrict__ scale,
                          const float* __restrict__ shift, float* __restrict__ out,
                          size_t total) {
  size_t i = (size_t)blockIdx.x * blockDim.x + threadIdx.x;
  if (i < total) {
    int c = (int)(i & (DM - 1));
    out[i] = z[i] * scale[c] + shift[c];
  }
}

__global__ __launch_bounds__(256) void k_pool(
    const float* __restrict__ h, const float* __restrict__ att,
    const int* __restrict__ gid, float* __restrict__ gsum,
    float* __restrict__ gcnt, int n) {
  int w    = (blockIdx.x * blockDim.x + threadIdx.x) >> 5;
  int lane = threadIdx.x & 31;
  if (w >= n) return;
  int   g = gid[w];
  float a = att[w];
  const float4 v = *(const float4*)(h + (size_t)w * DM + lane * 4);
  float* gp = gsum + (size_t)g * DM + lane * 4;
  atomicAdd(gp + 0, v.x * a);
  atomicAdd(gp + 1, v.y * a);
  atomicAdd(gp + 2, v.z * a);
  atomicAdd(gp + 3, v.w * a);
  if (lane == 0) atomicAdd(&gcnt[g], 1.f);
}

__global__ void k_finalize(const float* __restrict__ gsum, const float* __restrict__ gcnt,
                           const float* __restrict__ att, float* __restrict__ out) {
  int i = blockIdx.x * blockDim.x + threadIdx.x;
  const int ge = NG * DM;
  if (i < ge) out[i] = gsum[i] / fmaxf(gcnt[i / DM], 1.f);
  else if (i < ge + NN) out[i] = att[i - ge];
}

// ------------------------------- host side ----------------------------------
static void build_csr(const int* src, const int* dst, int E, const int* cnt_dst,
                      int* rowptr, int* cur, int* entries, int* bsumbuf,
                      hipStream_t stream) {
  int nb = (NN + 255) / 256;
  k_scan_block<<<nb, 256, 0, stream>>>(cnt_dst, rowptr, bsumbuf, NN);
  k_scan_single<<<1, 1024, 0, stream>>>(bsumbuf, nb);
  k_scan_add<<<nb, 256, 0, stream>>>(rowptr, bsumbuf, cur, NN);
  k_csr_fill<<<(E + 255) / 256, 256, 0, stream>>>(src, dst, cur, entries, E);
}

static void run_layer(const float* X, int K, const float* Wgc, const float* bgc,
                      const float* Wgat, const float* al, const float* ar,
                      const float* bgat, const float* g, const float* be, int H,
                      int act, float* A, float* B, float* C, float* el, float* er,
                      const float* doutinv, const float* dininv, const int* rp_i,
                      const int* cn_i, const int* en_i, const int* rp_c,
                      const int* cn_c, const int* en_c, float* attsum,
                      float* bsum, float* bsq, float* bsc, float* bsh,
                      float* hout, hipStream_t stream) {
  const int gB = (NN + 127) / 128;
  const int wB = (NN * 32 + 255) / 256;
  if (K == 512) k_gemm_wmma<512><<<gB, 256, 0, stream>>>(X, Wgc, A, doutinv, NN);
  else          k_gemm_wmma<128><<<gB, 256, 0, stream>>>(X, Wgc, A, doutinv, NN);
  k_gc_gather<<<wB, 256, 0, stream>>>(A, rp_i, cn_i, en_i, B, NN);
  if (K == 512) k_gemm_wmma<512><<<gB, 256, 0, stream>>>(X, Wgat, A, nullptr, NN);
  else          k_gemm_wmma<128><<<gB, 256, 0, stream>>>(X, Wgat, A, nullptr, NN);
  if (H == 4) {
    k_eler<4><<<(NN * 4 + 255) / 256, 256, 0, stream>>>(A, al, ar, el, er, NN);
    k_gat_gather<4><<<wB, 256, 0, stream>>>(A, el, er, rp_c, cn_c, en_c, C, nullptr, NN);
  } else {
    k_eler<1><<<(NN + 255) / 256, 256, 0, stream>>>(A, al, ar, el, er, NN);
    k_gat_gather<1><<<wB, 256, 0, stream>>>(A, el, er, rp_c, cn_c, en_c, C, attsum, NN);
  }
  (void)hipMemsetAsync(bsum, 0, DM * sizeof(float), stream);
  (void)hipMemsetAsync(bsq,  0, DM * sizeof(float), stream);
  k_combine<<<2048, 128, 0, stream>>>(B, dininv, bgc, C, bgat, B, bsum, bsq, NN, act);
  k_bnfinal<<<1, 128, 0, stream>>>(bsum, bsq, g, be, bsc, bsh);
  size_t total = (size_t)NN * DM;
  k_bnapply<<<(int)((total + 255) / 256), 256, 0, stream>>>(B, bsc, bsh, hout, total);
}

extern "C" void kernel_launch(void* const* d_in, const int* in_sizes, int n_in,
                              void* d_out, int out_size, void* d_ws, size_t ws_size,
                              hipStream_t stream) {
  const float* feat = (const float*)d_in[0];
  const int*   isrc = (const int*)d_in[1];
  const int*   idst = (const int*)d_in[2];
  const int*   csrc = (const int*)d_in[3];
  const int*   cdst = (const int*)d_in[4];
  const int*   gid  = (const int*)d_in[5];
  const float *Wgc1 = (const float*)d_in[6],  *bgc1 = (const float*)d_in[7];
  const float *Wga1 = (const float*)d_in[8],  *al1 = (const float*)d_in[9];
  const float *ar1  = (const float*)d_in[10], *bga1 = (const float*)d_in[11];
  const float *Wgc2 = (const float*)d_in[12], *bgc2 = (const float*)d_in[13];
  const float *Wga2 = (const float*)d_in[14], *al2 = (const float*)d_in[15];
  const float *ar2  = (const float*)d_in[16], *bga2 = (const float*)d_in[17];
  const float *Wgc3 = (const float*)d_in[18], *bgc3 = (const float*)d_in[19];
  const float *Wga3 = (const float*)d_in[20], *al3 = (const float*)d_in[21];
  const float *ar3  = (const float*)d_in[22], *bga3 = (const float*)d_in[23];
  const float *g1 = (const float*)d_in[24], *be1 = (const float*)d_in[25];
  const float *g2 = (const float*)d_in[26], *be2 = (const float*)d_in[27];
  const float *g3 = (const float*)d_in[28], *be3 = (const float*)d_in[29];
  float* out = (float*)d_out;

  char* p = (char*)d_ws;
  auto carve = [&](size_t bytes) -> char* {
    char* r = p;
    p += (bytes + 255) & ~(size_t)255;
    return r;
  };
  float* h    = (float*)carve(sizeof(float) * (size_t)NN * DM);
  float* A    = (float*)carve(sizeof(float) * (size_t)NN * DM);
  float* B    = (float*)carve(sizeof(float) * (size_t)NN * DM);
  float* C    = (float*)carve(sizeof(float) * (size_t)NN * DM);
  float* el   = (float*)carve(sizeof(float) * (size_t)NN * 4);
  float* er   = (float*)carve(sizeof(float) * (size_t)NN * 4);
  float* doutinv = (float*)carve(sizeof(float) * NN);
  float* dininv  = (float*)carve(sizeof(float) * NN);
  float* att     = (float*)carve(sizeof(float) * NN);
  float* bsum = (float*)carve(sizeof(float) * DM);
  float* bsq  = (float*)carve(sizeof(float) * DM);
  float* bsc  = (float*)carve(sizeof(float) * DM);
  float* bsh  = (float*)carve(sizeof(float) * DM);
  float* gsum = (float*)carve(sizeof(float) * (size_t)NG * DM);
  float* gcnt = (float*)carve(sizeof(float) * NG);
  int* cnt_src_i = (int*)carve(sizeof(int) * NN);
  int* cnt_dst_i = (int*)carve(sizeof(int) * NN);
  int* cnt_dst_c = (int*)carve(sizeof(int) * NN);
  int* rp_i = (int*)carve(sizeof(int) * NN);
  int* cur_i = (int*)carve(sizeof(int) * NN);
  int* en_i = (int*)carve(sizeof(int) * (size_t)NEI);
  int* rp_c = (int*)carve(sizeof(int) * NN);
  int* cur_c = (int*)carve(sizeof(int) * NN);
  int* en_c = (int*)carve(sizeof(int) * (size_t)NEC);
  int* bsumbuf = (int*)carve(sizeof(int) * 1024);

  // --- fresh per-call state ---
  (void)hipMemsetAsync(cnt_src_i, 0, sizeof(int) * NN, stream);
  (void)hipMemsetAsync(cnt_dst_i, 0, sizeof(int) * NN, stream);
  (void)hipMemsetAsync(cnt_dst_c, 0, sizeof(int) * NN, stream);
  (void)hipMemsetAsync(att, 0, sizeof(float) * NN, stream);
  (void)hipMemsetAsync(gsum, 0, sizeof(float) * (size_t)NG * DM, stream);
  (void)hipMemsetAsync(gcnt, 0, sizeof(float) * NG, stream);

  // --- degrees + CSR (by dst) for both edge sets ---
  k_count<<<(NEI + 255) / 256, 256, 0, stream>>>(isrc, idst, cnt_src_i, cnt_dst_i, NEI);
  k_count<<<(NEC + 255) / 256, 256, 0, stream>>>(csrc, cdst, nullptr, cnt_dst_c, NEC);
  k_deginv<<<(NN + 255) / 256, 256, 0, stream>>>(cnt_src_i, doutinv, NN);
  k_deginv<<<(NN + 255) / 256, 256, 0, stream>>>(cnt_dst_i, dininv, NN);
  build_csr(isrc, idst, NEI, cnt_dst_i, rp_i, cur_i, en_i, bsumbuf, stream);
  build_csr(csrc, cdst, NEC, cnt_dst_c, rp_c, cur_c, en_c, bsumbuf, stream);

  // --- layers ---
  run_layer(feat, 512, Wgc1, bgc1, Wga1, al1, ar1, bga1, g1, be1, 4, 1,
            A, B, C, el, er, doutinv, dininv, rp_i, cnt_dst_i, en_i,
            rp_c, cnt_dst_c, en_c, nullptr, bsum, bsq, bsc, bsh, h, stream);
  run_layer(h, 128, Wgc2, bgc2, Wga2, al2, ar2, bga2, g2, be2, 4, 1,
            A, B, C, el, er, doutinv, dininv, rp_i, cnt_dst_i, en_i,
            rp_c, cnt_dst_c, en_c, nullptr, bsum, bsq, bsc, bsh, h, stream);
  run_layer(h, 128, Wgc3, bgc3, Wga3, al3, ar3, bga3, g3, be3, 1, 0,
            A, B, C, el, er, doutinv, dininv, rp_i, cnt_dst_i, en_i,
            rp_c, cnt_dst_c, en_c, att, bsum, bsq, bsc, bsh, h, stream);

  // --- attention-weighted graph pooling ---
  k_pool<<<(NN * 32 + 255) / 256, 256, 0, stream>>>(h, att, gid, gsum, gcnt, NN);
  k_finalize<<<(NG * DM + NN + 255) / 256, 256, 0, stream>>>(gsum, gcnt, att, out);
}